// MOEnorm_9732395892915
// MI455X (gfx1250) — compile-verified
//
#include <hip/hip_runtime.h>
#include <hip/hip_bf16.h>

// Shapes fixed by the reference: N=64, C=256, H=W=56 (HW=3136=784 float4), K=8 experts.
#define N_ 64
#define C_ 256
#define K_ 8
#define HW_ 3136
#define HW4_ 784
#define EPS_ 1e-5f

typedef __attribute__((ext_vector_type(2))) float v2f;
typedef __attribute__((ext_vector_type(4))) float v4f;
typedef __attribute__((ext_vector_type(8))) float v8f;

// ---------------------------------------------------------------------------
// Kernel 1: per-(n,c) spatial mean of x and x^2. One block per plane.
// b128 streaming loads; wave32 shuffle reduction + 8-entry LDS combine.
// Loads use default (RT) temporal hint so x stays resident in the 192MB L2
// for the apply pass.
// ---------------------------------------------------------------------------
__global__ __launch_bounds__(256) void moe_stats_kernel(const float* __restrict__ x,
                                                        float* __restrict__ xm,
                                                        float* __restrict__ x2m) {
    __shared__ float s_sum[8];
    __shared__ float s_sq[8];
    const int p = blockIdx.x;                 // plane index = n*C + c
    const int tid = threadIdx.x;
    const int wave = tid >> 5;
    const int lane = tid & 31;
    const v4f* xv = reinterpret_cast<const v4f*>(x + (size_t)p * HW_);

    float sum = 0.f, sq = 0.f;
    for (int i = tid; i < HW4_; i += 256) {
        v4f v = xv[i];
        sum += v.x + v.y + v.z + v.w;
        sq  += v.x * v.x + v.y * v.y + v.z * v.z + v.w * v.w;
    }
    // wave32 butterfly reduce
    for (int off = 16; off > 0; off >>= 1) {
        sum += __shfl_down(sum, off, 32);
        sq  += __shfl_down(sq,  off, 32);
    }
    if (lane == 0) { s_sum[wave] = sum; s_sq[wave] = sq; }
    __syncthreads();
    if (tid == 0) {
        float ts = 0.f, tq = 0.f;
        for (int w = 0; w < 8; ++w) { ts += s_sum[w]; tq += s_sq[w]; }
        const float inv = 1.f / (float)HW_;
        xm[p]  = ts * inv;
        x2m[p] = tq * inv;
    }
}

// ---------------------------------------------------------------------------
// Kernel 2: gate softmax + expert stats + per-(n,c) affine coefficients.
// Single 256-thread block = 8 wave32s. GEMMs via V_WMMA_F32_16X16X4_F32.
//   mu/ex2 = gk[8(pad16),64] @ {xm,x2m}[64,256]     (32 tile jobs x 16 k-steps)
//   A/B    = g[64,8]         @ {rstd,mu*rstd}[8,256] (128 tile jobs x 2 k-steps)
// A-fragments come from a zero-padded transposed gate tile in LDS so every
// lane issues an unconditional ds_load (EXEC stays all-1s, no saveexec churn).
// WMMA calls sit under per-wave-uniform branches only.
// ---------------------------------------------------------------------------
__global__ __launch_bounds__(256) void moe_gate_wmma_kernel(const float* __restrict__ xm,
                                                            const float* __restrict__ x2m,
                                                            const float* __restrict__ Wg,
                                                            const float* __restrict__ Wb,
                                                            float* __restrict__ Amat,
                                                            float* __restrict__ Bmat) {
    __shared__ float g_lds[N_ * K_];      // gates g[n][k]
    __shared__ float gpadT[N_ * 16];      // gpadT[n][m] = (m<8) ? g[n][m] : 0  (A of phase B)
    __shared__ float invnk[K_];
    __shared__ float mu_l[K_ * C_];       // already scaled by inv_nk
    __shared__ float ex2_l[K_ * C_];      // already scaled by inv_nk
    __shared__ float rstd_l[K_ * C_];
    __shared__ float mrs_l[K_ * C_];      // mu * rstd

    const int tid = threadIdx.x;
    const int wave = tid >> 5;
    const int lane = tid & 31;
    const int lhalf = lane & 15;
    const int hi2 = (lane >> 4) & 1;      // 0: lanes 0-15, 1: lanes 16-31

    // ---- Phase A: logits = xm @ Wg^T + Wb; softmax over K=8 ----
    if (tid < N_) {
        float lg[K_];
        for (int k = 0; k < K_; ++k) lg[k] = Wb[k];
        const float* xrow = xm + tid * C_;
        for (int c = 0; c < C_; ++c) {
            const float xv = xrow[c];
            for (int k = 0; k < K_; ++k) lg[k] += xv * Wg[k * C_ + c];
        }
        float mx = lg[0];
        for (int k = 1; k < K_; ++k) mx = fmaxf(mx, lg[k]);
        float s = 0.f;
        float e[K_];
        for (int k = 0; k < K_; ++k) { e[k] = __expf(lg[k] - mx); s += e[k]; }
        const float inv = 1.f / s;
        for (int k = 0; k < K_; ++k) g_lds[tid * K_ + k] = e[k] * inv;
    }
    __syncthreads();
    if (tid < K_) {
        float s = 0.f;
        for (int n = 0; n < N_; ++n) s += g_lds[n * K_ + tid];
        invnk[tid] = 1.f / s;
    }
    // zero-padded transpose-friendly copy: gpadT[n*16 + m]
    for (int i = tid; i < N_ * 16; i += 256) {
        const int n = i >> 4;
        const int m = i & 15;
        gpadT[i] = (m < K_) ? g_lds[n * K_ + m] : 0.f;
    }
    __syncthreads();

    // ---- Phase B: mu = gk @ xm, ex2 = gk @ x2m  (WMMA 16x16x4 f32) ----
    for (int jj = 0; jj < 4; ++jj) {
        const int j = wave + 8 * jj;                  // 0..31, uniform per wave
        const float* src = (j & 16) ? x2m : xm;       // B source [64,256]
        float* dstl = (j & 16) ? ex2_l : mu_l;
        const int c0 = (j & 15) * 16;
        v8f acc = {};
        for (int s = 0; s < 16; ++s) {
            const int ka = 4 * s + 2 * hi2;           // K indices ka, ka+1 (n-dim)
            v2f a, b;
            // A[m][kidx] = g[kidx][m] padded: unconditional LDS loads
            a.x = gpadT[ka * 16 + lhalf];
            a.y = gpadT[(ka + 1) * 16 + lhalf];
            const int col = c0 + lhalf;
            b.x = src[ka * C_ + col];
            b.y = src[(ka + 1) * C_ + col];
            acc = __builtin_amdgcn_wmma_f32_16x16x4_f32(false, a, false, b,
                                                        (short)0, acc, false, false);
        }
        // D rows 0..7 (experts) live in lanes 0-15 of VGPRs 0..7; scale by inv_nk.
        if (!hi2) {
            for (int r = 0; r < K_; ++r)
                dstl[r * C_ + c0 + lhalf] = acc[r] * invnk[r];
        }
    }
    __syncthreads();

    // ---- rstd = rsqrt(ex2 - mu^2 + eps); mrs = mu*rstd ----
    for (int i = tid; i < K_ * C_; i += 256) {
        const float m = mu_l[i];
        const float r = __frsqrt_rn(ex2_l[i] - m * m + EPS_);
        rstd_l[i] = r;
        mrs_l[i] = m * r;
    }
    __syncthreads();

    // ---- Phase C: Amat = g @ rstd, Bmat = g @ mrs  (WMMA 16x16x4 f32) ----
    for (int jj = 0; jj < 16; ++jj) {
        const int j = wave + 8 * jj;                  // 0..127, uniform per wave
        const float* S = (j & 64) ? mrs_l : rstd_l;   // [8,256]
        float* dst = (j & 64) ? Bmat : Amat;
        const int t = j & 63;
        const int mt = t >> 4;                        // M tile (rows of g)
        const int c0 = (t & 15) * 16;
        v8f acc = {};
        for (int s = 0; s < 2; ++s) {
            const int ka = 4 * s + 2 * hi2;           // K indices (expert dim, 0..7)
            v2f a, b;
            const int m = 16 * mt + lhalf;            // sample index n
            a.x = g_lds[m * K_ + ka];
            a.y = g_lds[m * K_ + ka + 1];
            const int col = c0 + lhalf;
            b.x = S[ka * C_ + col];
            b.y = S[(ka + 1) * C_ + col];
            acc = __builtin_amdgcn_wmma_f32_16x16x4_f32(false, a, false, b,
                                                        (short)0, acc, false, false);
        }
        for (int r = 0; r < 8; ++r) {
            const int n = 16 * mt + r + 8 * hi2;      // D layout: M=r (lo) / r+8 (hi)
            dst[n * C_ + c0 + lhalf] = acc[r];
        }
    }
}

// ---------------------------------------------------------------------------
// Kernel 3: out = x * (alpha*A) + (beta - alpha*B).
// Planes processed in REVERSE order (L2 holds the tail of x after kernel 1;
// 205.5MB x vs 192MB L2), and output stores are NON-TEMPORAL (native vector
// type for the builtin) so the write stream doesn't evict resident x lines.
// ---------------------------------------------------------------------------
__global__ __launch_bounds__(256) void moe_apply_kernel(const float* __restrict__ x,
                                                        const float* __restrict__ Amat,
                                                        const float* __restrict__ Bmat,
                                                        const float* __restrict__ alpha,
                                                        const float* __restrict__ beta,
                                                        float* __restrict__ out) {
    const int p = (int)(gridDim.x - 1u) - (int)blockIdx.x;  // reverse plane order
    const int c = p & (C_ - 1);
    const int tid = threadIdx.x;
    const float al = alpha[c];
    const float scale = al * Amat[p];
    const float shift = beta[c] - al * Bmat[p];

    const v4f* xv = reinterpret_cast<const v4f*>(x + (size_t)p * HW_);
    v4f* ov = reinterpret_cast<v4f*>(out + (size_t)p * HW_);
    for (int i = tid; i < HW4_; i += 256) {
        v4f v = xv[i];
        v4f o = v * scale + shift;                // vector fma
        __builtin_nontemporal_store(o, ov + i);   // NT b128: don't evict x from L2
    }
}

extern "C" void kernel_launch(void* const* d_in, const int* in_sizes, int n_in,
                              void* d_out, int out_size, void* d_ws, size_t ws_size,
                              hipStream_t stream) {
    const float* x     = (const float*)d_in[0];  // [64,256,56,56]
    const float* Wg    = (const float*)d_in[1];  // [8,256]
    const float* Wb    = (const float*)d_in[2];  // [8]
    const float* alpha = (const float*)d_in[3];  // [1,256,1,1]
    const float* beta  = (const float*)d_in[4];  // [1,256,1,1]
    float* out = (float*)d_out;

    // Workspace layout (floats): xm | x2m | Amat | Bmat, each N*C = 16384.
    float* ws   = (float*)d_ws;
    float* xm   = ws;
    float* x2m  = ws + 16384;
    float* Amat = ws + 2 * 16384;
    float* Bmat = ws + 3 * 16384;

    const int planes = N_ * C_;  // 16384

    moe_stats_kernel<<<planes, 256, 0, stream>>>(x, xm, x2m);
    moe_gate_wmma_kernel<<<1, 256, 0, stream>>>(xm, x2m, Wg, Wb, Amat, Bmat);
    moe_apply_kernel<<<planes, 256, 0, stream>>>(x, Amat, Bmat, alpha, beta, out);
}